// ThreeStateExplorer_45981919871401
// MI455X (gfx1250) — compile-verified
//
#include <hip/hip_runtime.h>

// ThreeStateExplorer trajectory integration, CDNA5 (gfx1250).
// Per-row hierarchical scan: k = cumsum(states==2), gather e0s[k]*speed,
// X = x0 + dt*cumsum(vel). Bandwidth-bound (~0.5 GB -> ~22us @ 23.3 TB/s).
// Uses gfx1250 async global->LDS loads (ASYNCcnt) to double-buffer the
// states stream against the serially dependent tile scan.

#define BLOCK 1024
#define WAVES 32              // 1024 / wave32
#define ITEMS 4
#define TILE  (BLOCK * ITEMS) // 4096 steps per tile
#define DT_F  0.1f

#if defined(__AMDGCN__) && __has_builtin(__builtin_amdgcn_global_load_async_to_lds_b128)
#define USE_ASYNC 1
#else
#define USE_ASYNC 0
#endif

typedef int v4i __attribute__((vector_size(16)));
typedef __attribute__((address_space(1))) v4i* as1_v4i_p;
typedef __attribute__((address_space(3))) v4i* as3_v4i_p;

__device__ __forceinline__ void async_copy16(const int* gsrc, int* ldst) {
#if USE_ASYNC
  // gfx1250: global_load_async_to_lds_b128 (tracked by ASYNCcnt)
  __builtin_amdgcn_global_load_async_to_lds_b128(
      (as1_v4i_p)gsrc, (as3_v4i_p)ldst, /*offset=*/0, /*cpol=*/0);
#else
  *(int4*)ldst = *(const int4*)gsrc;
#endif
}

__device__ __forceinline__ void wait_async_lds() {
#if defined(__AMDGCN__) && __has_builtin(__builtin_amdgcn_s_wait_asynccnt)
  __builtin_amdgcn_s_wait_asynccnt(0);
#elif defined(__AMDGCN__)
  asm volatile("s_wait_asynccnt 0" ::: "memory");
#endif
}

__device__ __forceinline__ int wscan_i(int v, int lane) {
#pragma unroll
  for (int d = 1; d < 32; d <<= 1) {
    int n = __shfl_up(v, d, 32);
    if (lane >= d) v += n;
  }
  return v;
}

__device__ __forceinline__ float wscan_f(float v, int lane) {
#pragma unroll
  for (int d = 1; d < 32; d <<= 1) {
    float n = __shfl_up(v, d, 32);
    if (lane >= d) v += n;
  }
  return v;
}

__global__ __launch_bounds__(BLOCK) void three_state_traj_kernel(
    const int*   __restrict__ states,  // (B,T)
    const float* __restrict__ e0s,     // (B,T+1,3)
    const float* __restrict__ sp0,     // (B,)
    const float* __restrict__ sp1,     // (B,)
    const float* __restrict__ x0,      // (B,3)
    float*       __restrict__ out,     // (B,T,3)
    int T)
{
  __shared__ int   sst[2][TILE];       // double-buffered async states tiles
  __shared__ int   skw[WAVES];         // per-wave tumble partials
  __shared__ float svx[WAVES], svy[WAVES], svz[WAVES];

  const int b    = blockIdx.x;
  const int tid  = threadIdx.x;
  const int lane = tid & 31;
  const int wav  = tid >> 5;

  const int*   st = states + (size_t)b * (size_t)T;
  const float* e0 = e0s + (size_t)b * (size_t)(T + 1) * 3;
  const float  s0 = sp0[b], s1 = sp1[b];
  const float  ox = x0[3 * b + 0], oy = x0[3 * b + 1], oz = x0[3 * b + 2];
  float* o = out + (size_t)b * (size_t)T * 3;

  int   kcar = 0;                      // tumbles so far (carry)
  float vcx = 0.f, vcy = 0.f, vcz = 0.f;  // velocity cumsum carry

  const int ntile = (T + TILE - 1) / TILE;
  int clampmax = T - ITEMS; if (clampmax < 0) clampmax = 0;

  // Prime: async-prefetch tile 0 into buffer 0.
  {
    int gb = tid * ITEMS; if (gb > clampmax) gb = clampmax;
    async_copy16(st + gb, &sst[0][tid * ITEMS]);
  }
  wait_async_lds();
  __syncthreads();

  int buf = 0;
  for (int tile = 0; tile < ntile; ++tile) {
    const int t0 = tile * TILE;

    // Kick off next tile's states prefetch (overlaps with this tile's scan).
    if (tile + 1 < ntile) {
      int gb = t0 + TILE + tid * ITEMS; if (gb > clampmax) gb = clampmax;
      async_copy16(st + gb, &sst[buf ^ 1][tid * ITEMS]);
    }

    const int base = t0 + tid * ITEMS;
    int s_[ITEMS];
#pragma unroll
    for (int i = 0; i < ITEMS; ++i) s_[i] = sst[buf][tid * ITEMS + i];

    // ---- scan 1: inclusive tumble count k ----
    int c[ITEMS];
    int run = 0;
#pragma unroll
    for (int i = 0; i < ITEMS; ++i) {
      const bool tb = (base + i < T) && (s_[i] == 2);
      run += tb ? 1 : 0;
      c[i] = run;
    }
    const int wincl = wscan_i(run, lane);
    if (lane == 31) skw[wav] = wincl;
    __syncthreads();
    if (tid < WAVES) skw[tid] = wscan_i(skw[tid], lane);
    __syncthreads();
    const int woff  = wav ? skw[wav - 1] : 0;
    const int ktile = skw[WAVES - 1];
    const int kbase = kcar + woff + (wincl - run);  // exclusive prefix for item 0

    // ---- gather orientation, build velocity ----
    float vx[ITEMS], vy[ITEMS], vz[ITEMS];
#pragma unroll
    for (int i = 0; i < ITEMS; ++i) {
      const int t = base + i;
      float fx = 0.f, fy = 0.f, fz = 0.f;
      if (t < T && s_[i] != 2) {
        const float spd = (s_[i] == 0) ? s0 : s1;
        const float* e  = e0 + (size_t)(kbase + c[i]) * 3;
        fx = e[0] * spd; fy = e[1] * spd; fz = e[2] * spd;
      }
      vx[i] = fx; vy[i] = fy; vz[i] = fz;
    }

    // ---- scan 2: inclusive cumsum of velocity (3 components) ----
    float rx = 0.f, ry = 0.f, rz = 0.f;
#pragma unroll
    for (int i = 0; i < ITEMS; ++i) {
      rx += vx[i]; vx[i] = rx;
      ry += vy[i]; vy[i] = ry;
      rz += vz[i]; vz[i] = rz;
    }
    const float wx = wscan_f(rx, lane);
    const float wy = wscan_f(ry, lane);
    const float wz = wscan_f(rz, lane);
    if (lane == 31) { svx[wav] = wx; svy[wav] = wy; svz[wav] = wz; }
    __syncthreads();
    if (tid < WAVES) {
      svx[tid] = wscan_f(svx[tid], lane);
      svy[tid] = wscan_f(svy[tid], lane);
      svz[tid] = wscan_f(svz[tid], lane);
    }
    __syncthreads();
    const float bx = vcx + (wav ? svx[wav - 1] : 0.f) + (wx - rx);
    const float by = vcy + (wav ? svy[wav - 1] : 0.f) + (wy - ry);
    const float bz = vcz + (wav ? svz[wav - 1] : 0.f) + (wz - rz);
    const float ttx = svx[WAVES - 1], tty = svy[WAVES - 1], ttz = svz[WAVES - 1];

    // ---- write positions (3 x float4 per thread; aligned for full tiles) ----
    if (base + ITEMS <= T) {
      float r[12];
#pragma unroll
      for (int i = 0; i < ITEMS; ++i) {
        r[3 * i + 0] = ox + DT_F * (bx + vx[i]);
        r[3 * i + 1] = oy + DT_F * (by + vy[i]);
        r[3 * i + 2] = oz + DT_F * (bz + vz[i]);
      }
      float4* p = (float4*)(o + (size_t)base * 3);
      p[0] = make_float4(r[0], r[1], r[2],  r[3]);
      p[1] = make_float4(r[4], r[5], r[6],  r[7]);
      p[2] = make_float4(r[8], r[9], r[10], r[11]);
    } else {
#pragma unroll
      for (int i = 0; i < ITEMS; ++i) {
        const int t = base + i;
        if (t < T) {
          o[(size_t)t * 3 + 0] = ox + DT_F * (bx + vx[i]);
          o[(size_t)t * 3 + 1] = oy + DT_F * (by + vy[i]);
          o[(size_t)t * 3 + 2] = oz + DT_F * (bz + vz[i]);
        }
      }
    }

    // carries for next tile
    kcar += ktile;
    vcx += ttx; vcy += tty; vcz += ttz;

    // Drain this wave's async prefetch, publish LDS to all waves, swap buffers.
    wait_async_lds();
    __syncthreads();
    buf ^= 1;
  }
}

extern "C" void kernel_launch(void* const* d_in, const int* in_sizes, int n_in,
                              void* d_out, int out_size, void* d_ws, size_t ws_size,
                              hipStream_t stream) {
  const int*   states = (const int*)d_in[0];
  const float* e0s    = (const float*)d_in[1];
  const float* sp0    = (const float*)d_in[2];
  const float* sp1    = (const float*)d_in[3];
  const float* x0     = (const float*)d_in[4];
  float* out = (float*)d_out;

  const int B = in_sizes[2];            // speed_0 has B elements
  const int T = in_sizes[0] / B;        // states is (B,T)

  three_state_traj_kernel<<<B, BLOCK, 0, stream>>>(states, e0s, sp0, sp1, x0, out, T);
}